// HATencoder_9869834846897
// MI455X (gfx1250) — compile-verified
//
#include <hip/hip_runtime.h>
#include <hip/hip_bf16.h>
#include <math.h>

// ---------------------------------------------------------------------------
// HAT encoder on MI455X (gfx1250, wave32).
// fp32 everywhere (precision-sensitive arctanh/softmax chain; GEMMs are a
// tiny fraction of the byte traffic, so low-precision WMMA buys nothing).
// GEMMs use V_WMMA_F32_16X16X4_F32 on the matrix pipes.
// ---------------------------------------------------------------------------

typedef float v2f __attribute__((ext_vector_type(2)));
typedef float v8f __attribute__((ext_vector_type(8)));

#define EPSN 1e-7f

// ---------------- logmap0 row scale: s[n] = atanh(min(||x||,1-1e-5))/max(||x||,eps)
__global__ __launch_bounds__(256) void hat_logmap_scale(
    const float* __restrict__ X, float* __restrict__ S, int N, int K)
{
    int row  = blockIdx.x * (blockDim.x >> 5) + (threadIdx.x >> 5);
    int lane = threadIdx.x & 31;
    if (row >= N) return;
    const float* r = X + (size_t)row * K;
    float ss = 0.f;
    for (int k = lane; k < K; k += 32) { float v = r[k]; ss += v * v; }
    #pragma unroll
    for (int m = 16; m >= 1; m >>= 1) ss += __shfl_xor(ss, m, 32);
    if (lane == 0) {
        float n  = fmaxf(sqrtf(ss), EPSN);
        float nc = fminf(n, 1.0f - 1e-5f);
        S[row] = atanhf(nc) / n;
    }
}

// ---------------- fp32 WMMA GEMM: H[m,c] = sum_k (S[m]*X[m,k]) * W[k,c]
// One wave computes one 16x16 output tile; K stepped by 4 per WMMA.
// A layout (16x4 f32): lane = l16 + 16*g holds A[l16][k0+2g], A[l16][k0+2g+1].
// B layout (4x16 f32): lane = l16 + 16*g holds B[k0+2g][l16], B[k0+2g+1][l16].
// C/D layout: VGPR j, lane = n16 + 16*g -> D[j + 8*g][n16].
__global__ __launch_bounds__(32) void hat_gemm_wmma_f32(
    const float* __restrict__ X, const float* __restrict__ S,
    const float* __restrict__ W, float* __restrict__ Hout,
    int M, int K, int Ncols)
{
    const int lane = threadIdx.x & 31;
    const int g    = lane >> 4;
    const int l16  = lane & 15;
    const int row0 = blockIdx.x * 16;
    const int col0 = blockIdx.y * 16;

    const int   arow = row0 + l16;
    const int   crow = (arow < M) ? arow : (M - 1);
    const float sc   = (arow < M) ? S[arow] : 0.f;
    const float* __restrict__ xrow = X + (size_t)crow * K;
    const int   bcol = col0 + l16;

    v8f acc = {};
    #pragma unroll 4
    for (int k0 = 0; k0 < K; k0 += 4) {
        v2f a, b;
        a.x = xrow[k0 + 2 * g    ] * sc;
        a.y = xrow[k0 + 2 * g + 1] * sc;
        b.x = W[(size_t)(k0 + 2 * g    ) * Ncols + bcol];
        b.y = W[(size_t)(k0 + 2 * g + 1) * Ncols + bcol];
        // D = A*B + C, full fp32 on the matrix unit
        acc = __builtin_amdgcn_wmma_f32_16x16x4_f32(
            false, a, false, b, (short)0, acc, false, false);
    }
    #pragma unroll
    for (int j = 0; j < 8; ++j) {
        int r = row0 + j + 8 * g;
        if (r < M) Hout[(size_t)r * Ncols + (col0 + l16)] = acc[j];
    }
}

// ---------------- per-(node,head) attention projections
__global__ __launch_bounds__(256) void hat_attn_scores(
    const float* __restrict__ Hm, const float* __restrict__ a_src,
    const float* __restrict__ a_dst, float* __restrict__ ES,
    float* __restrict__ ED, int N, int H, int D)
{
    int idx = blockIdx.x * blockDim.x + threadIdx.x;
    if (idx >= N * H) return;
    int n = idx / H, h = idx % H;
    const float* hv = Hm + (size_t)n * H * D + (size_t)h * D;
    const float* as = a_src + h * D;
    const float* ad = a_dst + h * D;
    float s = 0.f, d = 0.f;
    for (int i = 0; i < D; ++i) { float v = hv[i]; s += v * as[i]; d += v * ad[i]; }
    ES[idx] = s;
    ED[idx] = d;
}

// ---------------- monotone float<->uint key for atomic segment-max
__device__ __forceinline__ unsigned fkey(float f) {
    unsigned u = __float_as_uint(f);
    return (u & 0x80000000u) ? ~u : (u | 0x80000000u);
}
__device__ __forceinline__ float fkey_dec(unsigned k) {
    unsigned u = (k & 0x80000000u) ? (k & 0x7FFFFFFFu) : ~k;
    return __uint_as_float(u);
}
__device__ __forceinline__ float lrelu(float v) { return v > 0.f ? v : 0.2f * v; }

__global__ __launch_bounds__(256) void hat_edge_max(
    const float* __restrict__ ES, const float* __restrict__ ED,
    const int* __restrict__ src, const int* __restrict__ dst,
    unsigned* __restrict__ EMAXK, int E, int H)
{
    int idx = blockIdx.x * blockDim.x + threadIdx.x;
    if (idx >= E * H) return;
    int e = idx / H, h = idx % H;
    int s = src[e], d = dst[e];
    float val = lrelu(ES[s * H + h] + ED[d * H + h]);
    atomicMax(&EMAXK[d * H + h], fkey(val));
}

// decode keys in place; unreachable segments (key==0 -> NaN) become 0
__global__ __launch_bounds__(256) void hat_emax_decode(
    unsigned* __restrict__ EMAXK, int NH)
{
    int idx = blockIdx.x * blockDim.x + threadIdx.x;
    if (idx >= NH) return;
    float f = fkey_dec(EMAXK[idx]);
    if (!isfinite(f)) f = 0.f;
    ((float*)EMAXK)[idx] = f;
}

__global__ __launch_bounds__(256) void hat_edge_exp_sum(
    const float* __restrict__ ES, const float* __restrict__ ED,
    const int* __restrict__ src, const int* __restrict__ dst,
    const float* __restrict__ EMAX, float* __restrict__ EE,
    float* __restrict__ DEN, int E, int H)
{
    int idx = blockIdx.x * blockDim.x + threadIdx.x;
    if (idx >= E * H) return;
    int e = idx / H, h = idx % H;
    int s = src[e], d = dst[e];
    float val = lrelu(ES[s * H + h] + ED[d * H + h]);
    float ee  = __expf(val - EMAX[d * H + h]);
    EE[idx] = ee;
    atomicAdd(&DEN[d * H + h], ee);
}

// ---------------- out[dst] += alpha * h[src]; thread = (edge, 4-float chunk)
__global__ __launch_bounds__(256) void hat_edge_aggregate(
    const float* __restrict__ Hm, const int* __restrict__ src,
    const int* __restrict__ dst, const float* __restrict__ EE,
    const float* __restrict__ DEN, float* __restrict__ AGG,
    int E, int H, int D)
{
    const int chunks = (H * D) >> 2;          // float4 chunks per edge
    int idx = blockIdx.x * blockDim.x + threadIdx.x;
    if (idx >= E * chunks) return;
    int e = idx / chunks, j = idx % chunks;
    int head = (j * 4) / D;
    int s = src[e], d = dst[e];
    float alpha = EE[e * H + head] / (DEN[d * H + head] + 1e-16f);
    float4 hv = ((const float4*)(Hm + (size_t)s * H * D))[j];
    float* ob = AGG + (size_t)d * H * D + (size_t)j * 4;
    atomicAdd(ob + 0, alpha * hv.x);
    atomicAdd(ob + 1, alpha * hv.y);
    atomicAdd(ob + 2, alpha * hv.z);
    atomicAdd(ob + 3, alpha * hv.w);
}

// ---------------- expmap0: row *= tanh(||row||)/max(||row||,eps)
__global__ __launch_bounds__(256) void hat_expmap_rows(
    const float* __restrict__ in, float* __restrict__ out, int N, int C)
{
    int row  = blockIdx.x * (blockDim.x >> 5) + (threadIdx.x >> 5);
    int lane = threadIdx.x & 31;
    if (row >= N) return;
    const float* r = in + (size_t)row * C;
    float ss = 0.f;
    for (int k = lane; k < C; k += 32) { float v = r[k]; ss += v * v; }
    #pragma unroll
    for (int m = 16; m >= 1; m >>= 1) ss += __shfl_xor(ss, m, 32);
    float n = fmaxf(sqrtf(ss), EPSN);
    float f = tanhf(n) / n;
    float* o = out + (size_t)row * C;
    for (int k = lane; k < C; k += 32) o[k] = r[k] * f;
}

// ---------------------------------------------------------------------------
static inline int cdiv(long long a, long long b) { return (int)((a + b - 1) / b); }

extern "C" void kernel_launch(void* const* d_in, const int* in_sizes, int n_in,
                              void* d_out, int out_size, void* d_ws, size_t ws_size,
                              hipStream_t stream)
{
    (void)n_in; (void)out_size; (void)ws_size;
    const float* x      = (const float*)d_in[0];
    const float* W1     = (const float*)d_in[1];
    const float* a_src1 = (const float*)d_in[2];
    const float* a_dst1 = (const float*)d_in[3];
    const float* W2     = (const float*)d_in[4];
    const float* a_src2 = (const float*)d_in[5];
    const float* a_dst2 = (const float*)d_in[6];
    const int*   ei0    = (const int*)d_in[7];
    const int*   ei1    = (const int*)d_in[8];
    float*       out    = (float*)d_out;

    const int IN = 128, H1 = 4, D1 = 64, C1 = H1 * D1 /*256*/, H2 = 1, D2 = 64;
    const int N = in_sizes[0] / IN;
    const int E = in_sizes[7] / 2;
    const int *src0 = ei0, *dst0 = ei0 + E;
    const int *src1 = ei1, *dst1 = ei1 + E;

    // workspace layout (fp32), regions 1KB-aligned
    auto align_f = [](size_t f) { return (f + 255) & ~(size_t)255; };
    float* base = (float*)d_ws;
    size_t off = 0;
    float* S    = base + off; off += align_f((size_t)N);
    float* HBUF = base + off; off += align_f((size_t)N * C1);   // layer1 h / layer2 h
    float* AGG  = base + off; off += align_f((size_t)N * C1);   // agg / x2 / agg2
    float* ES   = base + off; off += align_f((size_t)N * H1);
    float* ED   = base + off; off += align_f((size_t)N * H1);
    unsigned* EMAX = (unsigned*)(base + off); off += align_f((size_t)N * H1);
    float* DEN  = base + off; off += align_f((size_t)N * H1);
    float* EE   = base + off; off += align_f((size_t)E * H1);

    // ======================= layer 1 (H=4, D=64) =======================
    hat_logmap_scale<<<cdiv(N, 8), 256, 0, stream>>>(x, S, N, IN);
    hat_gemm_wmma_f32<<<dim3(cdiv(N, 16), C1 / 16), 32, 0, stream>>>(
        x, S, W1, HBUF, N, IN, C1);
    hat_attn_scores<<<cdiv((long long)N * H1, 256), 256, 0, stream>>>(
        HBUF, a_src1, a_dst1, ES, ED, N, H1, D1);
    hipMemsetAsync(EMAX, 0, (size_t)N * H1 * 4, stream);
    hipMemsetAsync(DEN,  0, (size_t)N * H1 * 4, stream);
    hipMemsetAsync(AGG,  0, (size_t)N * C1 * 4, stream);
    hat_edge_max<<<cdiv((long long)E * H1, 256), 256, 0, stream>>>(
        ES, ED, src0, dst0, EMAX, E, H1);
    hat_emax_decode<<<cdiv((long long)N * H1, 256), 256, 0, stream>>>(EMAX, N * H1);
    hat_edge_exp_sum<<<cdiv((long long)E * H1, 256), 256, 0, stream>>>(
        ES, ED, src0, dst0, (const float*)EMAX, EE, DEN, E, H1);
    hat_edge_aggregate<<<cdiv((long long)E * (C1 / 4), 256), 256, 0, stream>>>(
        HBUF, src0, dst0, EE, DEN, AGG, E, H1, D1);
    hat_expmap_rows<<<cdiv(N, 8), 256, 0, stream>>>(AGG, AGG, N, C1); // x2 in place

    // ======================= layer 2 (H=1, D=64) =======================
    hat_logmap_scale<<<cdiv(N, 8), 256, 0, stream>>>(AGG, S, N, C1);
    hat_gemm_wmma_f32<<<dim3(cdiv(N, 16), D2 / 16), 32, 0, stream>>>(
        AGG, S, W2, HBUF, N, C1, D2);                       // HBUF now [N,64]
    hat_attn_scores<<<cdiv((long long)N * H2, 256), 256, 0, stream>>>(
        HBUF, a_src2, a_dst2, ES, ED, N, H2, D2);
    hipMemsetAsync(EMAX, 0, (size_t)N * H2 * 4, stream);
    hipMemsetAsync(DEN,  0, (size_t)N * H2 * 4, stream);
    hipMemsetAsync(AGG,  0, (size_t)N * D2 * 4, stream);    // agg2 reuses AGG region
    hat_edge_max<<<cdiv((long long)E * H2, 256), 256, 0, stream>>>(
        ES, ED, src1, dst1, EMAX, E, H2);
    hat_emax_decode<<<cdiv((long long)N * H2, 256), 256, 0, stream>>>(EMAX, N * H2);
    hat_edge_exp_sum<<<cdiv((long long)E * H2, 256), 256, 0, stream>>>(
        ES, ED, src1, dst1, (const float*)EMAX, EE, DEN, E, H2);
    hat_edge_aggregate<<<cdiv((long long)E * (D2 / 4), 256), 256, 0, stream>>>(
        HBUF, src1, dst1, EE, DEN, AGG, E, H2, D2);
    hat_expmap_rows<<<cdiv(N, 8), 256, 0, stream>>>(AGG, out, N, D2);
}